// PeepholeBidiLSTM_5351529251323
// MI455X (gfx1250) — compile-verified
//
#include <hip/hip_runtime.h>
#include <hip/hip_bf16.h>
#include <math.h>

// ---------------------------------------------------------------------------
// Persistent bidirectional peephole-LSTM for MI455X (gfx1250, wave32).
//
//  grid = 64 WGs (32 fwd + 32 rev), block = 256 threads (8 wave32 waves).
//  WG w of direction d owns h-columns [w*16, w*16+16):
//    - W_hh slice (64 gate cols x 512 K, fp32, +4-word col pad) lives in LDS
//    - per-step GEMM: gates[32,64] += A[32,K] * W^T  via v_wmma_f32_16x16x4_f32
//      (K = 512 x-part from L2-resident W_ih + 512 h-part from LDS)
//    - 4 independent WMMA accumulators break the C-dependence chain so the
//      SIMD can keep 4 matrix ops in flight instead of stalling on latency
//    - c-state stays in LDS; h exchanged via global double buffer + one
//      atomic arrive/spin grid barrier per timestep per direction.
// ---------------------------------------------------------------------------

#define LSTM_B 32
#define LSTM_T 2048
#define LSTM_I 512
#define LSTM_H 512
#define WG_PER_DIR 32
#define COLPAD 516              // 512 + 4 words: conflict-free LDS banks
#define W_LDS_FLOATS (64 * COLPAD)        // 33024
#define GATES_OFF W_LDS_FLOATS            // 2*4*256 floats
#define CSTATE_OFF (W_LDS_FLOATS + 2048)  // 512 floats
#define SMEM_FLOATS (W_LDS_FLOATS + 2048 + 512)

typedef __attribute__((ext_vector_type(2))) float v2f;
typedef __attribute__((ext_vector_type(8))) float v8f;

__global__ void lstm_zero_counters(int* cnt, int n) {
    int i = blockIdx.x * blockDim.x + threadIdx.x;
    if (i < n) cnt[i] = 0;
}

__device__ __forceinline__ float sigm(float v) {
    return 1.0f / (1.0f + expf(-v));
}

#define WMMA_F32(A, Bv, C) \
    __builtin_amdgcn_wmma_f32_16x16x4_f32(false, (A), false, (Bv), (short)0, (C), false, false)

__global__ __launch_bounds__(256, 1)
void lstm_persistent(const float* __restrict__ x,
                     const float* __restrict__ wih_f, const float* __restrict__ whh_f,
                     const float* __restrict__ wip_f, const float* __restrict__ wfp_f,
                     const float* __restrict__ wop_f,
                     const float* __restrict__ wih_r, const float* __restrict__ whh_r,
                     const float* __restrict__ wip_r, const float* __restrict__ wfp_r,
                     const float* __restrict__ wop_r,
                     float* __restrict__ out, float* __restrict__ hn, float* __restrict__ cn,
                     float* __restrict__ hbuf, int* __restrict__ cnt) {
    extern __shared__ float smem[];
    float* gatesL = smem + GATES_OFF;   // [2 rowtiles][4 gates][16][16]
    float* cstate = smem + CSTATE_OFF;  // [32][16]

    const int tid  = threadIdx.x;
    const int dir  = blockIdx.x >> 5;        // 0 = fwd, 1 = rev
    const int wg   = blockIdx.x & 31;
    const int j0   = wg * 16;                // owned h-column base

    const float* wih = dir ? wih_r : wih_f;
    const float* whh = dir ? whh_r : whh_f;
    const float* wip = dir ? wip_r : wip_f;
    const float* wfp = dir ? wfp_r : wfp_f;
    const float* wop = dir ? wop_r : wop_f;

    // ---- stage W_hh slice into LDS (64 gate cols x 512 K) -----------------
    for (int it = tid; it < 64 * 128; it += 256) {
        const int c  = it >> 7;              // 0..63 slice column
        const int k4 = (it & 127) << 2;      // 0..508 step 4
        const int gcol = (c >> 4) * LSTM_H + j0 + (c & 15);
        const float4 v = *(const float4*)(whh + (size_t)gcol * LSTM_H + k4);
        *(float4*)&smem[c * COLPAD + k4] = v;
    }
    // ---- zero cell state --------------------------------------------------
    cstate[tid]       = 0.0f;
    cstate[tid + 256] = 0.0f;
    __syncthreads();

    // ---- per-wave tile identity -------------------------------------------
    const int lane = tid & 31;
    const int wv   = tid >> 5;       // 0..7
    const int r    = wv & 1;         // batch row-tile (rows r*16..r*16+15)
    const int g    = wv >> 1;        // gate index 0..3
    const int half = lane >> 4;      // K sub-pair select
    const int ln   = lane & 15;
    const int kh   = half * 2;
    const int mA   = r * 16 + ln;                    // batch row fed to A
    const int gcolB = g * LSTM_H + j0 + ln;          // W row fed to B
    const float* wihB = wih + (size_t)gcolB * LSTM_I;
    const float* whhB = &smem[(g * 16 + ln) * COLPAD];

    const size_t hSlot = (size_t)LSTM_B * LSTM_H;
    float* hbase = hbuf + (size_t)dir * 2 * hSlot;
    int*   mycnt = cnt + dir * LSTM_T;
    const int dirOff = dir * LSTM_H;

    for (int s = 0; s < LSTM_T; ++s) {
        const int t = dir ? (LSTM_T - 1 - s) : s;

        // ---- GEMM phase: acc = x_t * W_ih^T + h_{s-1} * W_hh^T ------------
        // 4 independent accumulators (k mod 16) keep 4 WMMAs in flight.
        v8f ac0 = {}, ac1 = {}, ac2 = {}, ac3 = {};
        {
            const float* xrow = x + ((size_t)mA * LSTM_T + t) * LSTM_I;
            if (s + 1 < LSTM_T) {                     // prefetch next step's x
                const int tn = dir ? (t - 1) : (t + 1);
                const float* nx = x + ((size_t)mA * LSTM_T + tn) * LSTM_I;
                __builtin_prefetch(nx, 0, 1);
                __builtin_prefetch(nx + 256, 0, 1);
            }
#pragma unroll 2
            for (int kk = 0; kk < LSTM_I; kk += 16) {
                v2f a0 = *(const v2f*)(xrow + kk + kh);
                v2f b0 = *(const v2f*)(wihB + kk + kh);
                v2f a1 = *(const v2f*)(xrow + kk + 4 + kh);
                v2f b1 = *(const v2f*)(wihB + kk + 4 + kh);
                v2f a2 = *(const v2f*)(xrow + kk + 8 + kh);
                v2f b2 = *(const v2f*)(wihB + kk + 8 + kh);
                v2f a3 = *(const v2f*)(xrow + kk + 12 + kh);
                v2f b3 = *(const v2f*)(wihB + kk + 12 + kh);
                ac0 = WMMA_F32(a0, b0, ac0);
                ac1 = WMMA_F32(a1, b1, ac1);
                ac2 = WMMA_F32(a2, b2, ac2);
                ac3 = WMMA_F32(a3, b3, ac3);
            }
        }
        if (s > 0) {
            const float* hrow = hbase + ((s + 1) & 1) * hSlot + (size_t)mA * LSTM_H;
#pragma unroll 2
            for (int kk = 0; kk < LSTM_H; kk += 16) {
                v2f a0 = *(const v2f*)(hrow + kk + kh);
                v2f b0 = *(const v2f*)(whhB + kk + kh);
                v2f a1 = *(const v2f*)(hrow + kk + 4 + kh);
                v2f b1 = *(const v2f*)(whhB + kk + 4 + kh);
                v2f a2 = *(const v2f*)(hrow + kk + 8 + kh);
                v2f b2 = *(const v2f*)(whhB + kk + 8 + kh);
                v2f a3 = *(const v2f*)(hrow + kk + 12 + kh);
                v2f b3 = *(const v2f*)(whhB + kk + 12 + kh);
                ac0 = WMMA_F32(a0, b0, ac0);
                ac1 = WMMA_F32(a1, b1, ac1);
                ac2 = WMMA_F32(a2, b2, ac2);
                ac3 = WMMA_F32(a3, b3, ac3);
            }
        }
        const v8f acc = (ac0 + ac1) + (ac2 + ac3);

        // ---- scatter C tile to LDS (row M = vgpr + 8*laneHalf, col N = ln)
        {
            float* gt = gatesL + (r * 4 + g) * 256;
#pragma unroll
            for (int vv = 0; vv < 8; ++vv)
                gt[(vv + 8 * half) * 16 + ln] = acc[vv];
        }
        __syncthreads();

        // ---- elementwise LSTM cell over the owned 32x16 block -------------
        float* hw = hbase + (s & 1) * hSlot;
#pragma unroll
        for (int e = tid; e < 512; e += 256) {
            const int mrow = e >> 4;                 // batch row 0..31
            const int col  = e & 15;
            const int rr   = mrow >> 4, mr = mrow & 15;
            const float gi = gatesL[(rr * 4 + 0) * 256 + mr * 16 + col];
            const float gf = gatesL[(rr * 4 + 1) * 256 + mr * 16 + col];
            const float gg = gatesL[(rr * 4 + 2) * 256 + mr * 16 + col];
            const float go = gatesL[(rr * 4 + 3) * 256 + mr * 16 + col];
            const int   j  = j0 + col;
            const float c0 = cstate[e];
            const float ig = sigm(gi + wip[j] * c0);
            const float fg = sigm(gf + wfp[j] * c0);
            const float cy = fg * c0 + ig * tanhf(gg);
            const float og = go + wop[j] * cy;       // faithful: no sigmoid
            const float hy = og * tanhf(cy);
            cstate[e] = cy;
            hw[(size_t)mrow * LSTM_H + j] = hy;
            out[((size_t)mrow * LSTM_T + t) * (2 * LSTM_H) + dirOff + j] = hy;
            if (s == LSTM_T - 1) {
                hn[(size_t)dir * LSTM_B * LSTM_H + (size_t)mrow * LSTM_H + j] = hy;
                cn[(size_t)dir * LSTM_B * LSTM_H + (size_t)mrow * LSTM_H + j] = cy;
            }
        }

        // ---- per-direction grid barrier (32 WGs) --------------------------
        __threadfence();
        __syncthreads();
        if (tid == 0) {
            __hip_atomic_fetch_add(&mycnt[s], 1, __ATOMIC_ACQ_REL,
                                   __HIP_MEMORY_SCOPE_AGENT);
            while (__hip_atomic_load(&mycnt[s], __ATOMIC_ACQUIRE,
                                     __HIP_MEMORY_SCOPE_AGENT) < WG_PER_DIR) {
                __builtin_amdgcn_s_sleep(1);
            }
        }
        __syncthreads();
    }
}

extern "C" void kernel_launch(void* const* d_in, const int* in_sizes, int n_in,
                              void* d_out, int out_size, void* d_ws, size_t ws_size,
                              hipStream_t stream) {
    (void)in_sizes; (void)n_in; (void)out_size; (void)ws_size;
    const float* x     = (const float*)d_in[0];
    const float* wih_f = (const float*)d_in[1];
    const float* whh_f = (const float*)d_in[2];
    const float* wip_f = (const float*)d_in[3];
    const float* wfp_f = (const float*)d_in[4];
    const float* wop_f = (const float*)d_in[5];
    const float* wih_r = (const float*)d_in[6];
    const float* whh_r = (const float*)d_in[7];
    const float* wip_r = (const float*)d_in[8];
    const float* wfp_r = (const float*)d_in[9];
    const float* wop_r = (const float*)d_in[10];

    float* out = (float*)d_out;
    float* hn  = out + (size_t)LSTM_B * LSTM_T * 2 * LSTM_H;
    float* cn  = hn + 2 * LSTM_B * LSTM_H;

    int*   cnt  = (int*)d_ws;                                  // 2*T ints
    float* hbuf = (float*)((char*)d_ws + 2 * LSTM_T * sizeof(int));  // 256 KB

    lstm_zero_counters<<<(2 * LSTM_T + 255) / 256, 256, 0, stream>>>(cnt, 2 * LSTM_T);

    const size_t shmem = SMEM_FLOATS * sizeof(float);          // ~142 KB LDS
    lstm_persistent<<<dim3(2 * WG_PER_DIR), dim3(256), shmem, stream>>>(
        x, wih_f, whh_f, wip_f, wfp_f, wop_f,
        wih_r, whh_r, wip_r, wfp_r, wop_r,
        out, hn, cn, hbuf, cnt);
}